// GP_EP_CondRegLayer_51651276702561
// MI455X (gfx1250) — compile-verified
//
#include <hip/hip_runtime.h>

typedef __attribute__((ext_vector_type(2))) float v2f;
typedef __attribute__((ext_vector_type(8))) float v8f;

// Problem constants
#define N_TOT   2048            // 4*512 samples
#define D_DIM   32
#define C_DIM   8
#define P_DIM   32              // M_LEN
#define T_DIM   258             // 2 + C*M_LEN  (z_rem is empty)
#define E_DIM   34              // 2 + M_LEN (folded GEMM K-dim)
#define WROW    36              // E padded to multiple of 4 for WMMA K=4
#define ZOUT_ELEMS ((size_t)4 * 512 * 8 * 32 * 34)   // 17,825,792

// ---------------------------------------------------------------------------
// Kernel 1: per-(d,c) precompute. One wave32 per (d,c) pair (256 blocks).
// Builds W[32][36] = [A_mu | L | 0 0] and partial log-det, into workspace.
// ---------------------------------------------------------------------------
__global__ __launch_bounds__(32) void gp_precompute(
    const float* __restrict__ log_tau,
    const float* __restrict__ ts0,
    float* __restrict__ Wmat,
    float* __restrict__ parts) {
  const int dc = blockIdx.x;          // dc = d*8 + c
  const int d  = dc >> 3;
  const int c  = dc & 7;
  const int lane = threadIdx.x;       // 0..31, lane == row p

  __shared__ float S[32][33];         // Sigma -> in-place Cholesky (pad vs bank conflicts)
  __shared__ float kpS[32][2];        // K_predEP rows
  __shared__ float tsS[32];
  __shared__ float red[32];

  const float tau    = expf(log_tau[d]);
  const float inv2t2 = 1.0f / (2.0f * tau * tau);
  const float tp     = 20.0f + 2.0f * (float)c;   // TPS[c]

  const float t_p = ts0[lane];
  tsS[lane] = t_p;

  // K_predEP[p,0] = K(ts0[p], 0);  K_predEP[p,1] = K(ts0[p], TPS[c])
  const float kp0 = expf(-(t_p * t_p) * inv2t2);
  const float dte = t_p - tp;
  const float kp1 = expf(-(dte * dte) * inv2t2);
  kpS[lane][0] = kp0;
  kpS[lane][1] = kp1;

  // 2x2 EP block: [[1, b],[b, 1]], inv = [[1,-b],[-b,1]]/det
  const float b   = expf(-(tp * tp) * inv2t2);
  const float det = 1.0f - b * b;
  const float a0  = (kp0 - b * kp1) / det;   // A_mu[p,0]
  const float a1  = (kp1 - b * kp0) / det;   // A_mu[p,1]
  __syncthreads();

  // Sigma[p,q] = K(ts0[p], ts0[q]) - A_mu[p,:] . K_predEP[q,:]
  for (int q = 0; q < 32; ++q) {
    const float dd = t_p - tsS[q];
    S[lane][q] = expf(-(dd * dd) * inv2t2) - a0 * kpS[q][0] - a1 * kpS[q][1];
  }
  __syncthreads();

  // In-place lower Cholesky (right-looking), lane i owns row i.
  for (int j = 0; j < 32; ++j) {
    if (lane == j) S[j][j] = sqrtf(S[j][j]);
    __syncthreads();
    const float djj = S[j][j];
    if (lane > j) S[lane][j] = S[lane][j] / djj;
    __syncthreads();
    if (lane > j) {
      const float lij = S[lane][j];
      for (int k = j + 1; k <= lane; ++k) S[lane][k] -= lij * S[k][j];
    }
    __syncthreads();
  }

  // Partial log-det for this (d,c)
  red[lane] = logf(fabsf(S[lane][lane]));
  __syncthreads();
  if (lane == 0) {
    float s = 0.0f;
    for (int i = 0; i < 32; ++i) s += red[i];
    parts[dc] = s;
  }

  // Emit W row p: [a0, a1, L[p,0..31] (upper zeroed), 0, 0]
  float* wp = Wmat + ((size_t)dc * 32 + lane) * WROW;
  wp[0] = a0;
  wp[1] = a1;
  for (int q = 0; q < 32; ++q) wp[2 + q] = (q <= lane) ? S[lane][q] : 0.0f;
  wp[34] = 0.0f;
  wp[35] = 0.0f;
}

// ---------------------------------------------------------------------------
// Kernel 2: the batched GEMM via V_WMMA_F32_16X16X4_F32.
// One wave per (dc, 16-row block): 256 * 128 = 32768 wave-units, 8 waves/blk
// (all 8 waves of a block share one dc -> W rows hit in L2/L0).
// out[n, p] = sum_e x[n,e] * W[p,e]   (+ passthrough first/last columns)
// ---------------------------------------------------------------------------
__global__ __launch_bounds__(256) void gp_gemm(
    const float* __restrict__ z,
    const float* __restrict__ Wmat,
    float* __restrict__ out) {
  const int tid  = threadIdx.x;
  const int wave = tid >> 5;
  const int lane = tid & 31;
  const int unit = blockIdx.x * 8 + wave;      // 0..32767
  const int dc   = unit >> 7;                  // 0..255
  const int mb   = unit & 127;                 // 0..127 (16-row blocks)
  const int d    = dc >> 3;
  const int c    = dc & 7;
  const int n0   = mb << 4;

  const int half = lane >> 4;                  // K-half select (ISA A/B layout)
  const int l16  = lane & 15;                  // A: row M ; B: col N

  // This lane's A-row base: z[k,m,d,:] for n = n0 + l16
  const float* xrow  = z + ((size_t)(n0 + l16) * D_DIM + d) * T_DIM;
  const float* wbase = Wmat + (size_t)dc * 32 * WROW;
  const float* wr0   = wbase + (size_t)l16 * WROW;        // W rows p = l16
  const float* wr1   = wbase + (size_t)(16 + l16) * WROW; // W rows p = 16 + l16

  v8f acc0 = {};   // N = 0..15
  v8f acc1 = {};   // N = 16..31

  // K loop over e in [0,36) in steps of 4.  x(e): e<2 -> t=e ; 2<=e<34 -> t=c*32+e.
  // For e>=34 (pad lanes of the last step) we CLAMP the index to t=0 instead of
  // branching: W columns 34/35 are zero, so the product contributes nothing and
  // the load stays unconditional (no exec-mask divergence around WMMA).
  for (int s = 0; s < 9; ++s) {
    const int ea = s * 4 + (half << 1);
    const int eb = ea + 1;
    const int ta = (ea < 2) ? ea : ((ea < E_DIM) ? (c * 32 + ea) : 0);
    const int tb = (eb < 2) ? eb : ((eb < E_DIM) ? (c * 32 + eb) : 0);

    v2f a;  a.x  = xrow[ta]; a.y  = xrow[tb];
    v2f b0; b0.x = wr0[ea];  b0.y = wr0[eb];   // W padded with zeros at 34,35
    v2f b1; b1.x = wr1[ea];  b1.y = wr1[eb];

    acc0 = __builtin_amdgcn_wmma_f32_16x16x4_f32(
        false, a, false, b0, (short)0, acc0, false, false);
    acc1 = __builtin_amdgcn_wmma_f32_16x16x4_f32(
        false, a, false, b1, (short)0, acc1, false, false);
  }

  // Store z_hat into out[k,m,c,d,1+p]. C/D layout: VGPR r -> row r + 8*half,
  // lane%16 -> col. Columns are lane-contiguous -> coalesced stores.
  for (int r = 0; r < 8; ++r) {
    const int    n     = n0 + r + (half << 3);
    const size_t obase = (((size_t)n * C_DIM + c) * D_DIM + d) * 34;
    out[obase + 1 + l16]      = acc0[r];
    out[obase + 1 + 16 + l16] = acc1[r];
  }

  // Passthrough columns: col 0 = z[...,0] (lanes 0-15), col 33 = z[...,1] (16-31)
  {
    const int    n     = n0 + l16;
    const size_t obase = (((size_t)n * C_DIM + c) * D_DIM + d) * 34;
    const size_t zbase = ((size_t)n * D_DIM + d) * T_DIM;
    const float  v     = z[zbase + half];          // unconditional load
    out[obase + (half ? 33 : 0)] = v;              // address select, no branch
  }
}

// ---------------------------------------------------------------------------
// Kernel 3: sldj_out[n] = sldj_in[n] + sum(parts).  2048 threads.
// ---------------------------------------------------------------------------
__global__ __launch_bounds__(256) void gp_sldj(
    const float* __restrict__ sldj_in,
    const float* __restrict__ parts,
    float* __restrict__ out) {
  const int n = blockIdx.x * blockDim.x + threadIdx.x;   // 0..2047
  float tot = 0.0f;
  for (int i = 0; i < 256; ++i) tot += parts[i];
  out[ZOUT_ELEMS + n] = sldj_in[n] + tot;
}

// ---------------------------------------------------------------------------
extern "C" void kernel_launch(void* const* d_in, const int* in_sizes, int n_in,
                              void* d_out, int out_size, void* d_ws, size_t ws_size,
                              hipStream_t stream) {
  const float* z       = (const float*)d_in[0];   // (4,512,32,258)
  const float* log_tau = (const float*)d_in[1];   // (32,)
  const float* ts0     = (const float*)d_in[2];   // (32,)
  const float* sldj    = (const float*)d_in[3];   // (4,512)
  float* out = (float*)d_out;

  float* Wmat  = (float*)d_ws;                    // 256*32*36 floats (~1.13 MB)
  float* parts = Wmat + 256 * 32 * WROW;          // 256 floats

  gp_precompute<<<256, 32, 0, stream>>>(log_tau, ts0, Wmat, parts);
  gp_gemm<<<4096, 256, 0, stream>>>(z, Wmat, out);
  gp_sldj<<<8, 256, 0, stream>>>(sldj, parts, out);
}